// Deformer_88940182766004
// MI455X (gfx1250) — compile-verified
//
#include <hip/hip_runtime.h>
#include <stdint.h>

// Problem geometry (fixed by the reference).
#define B_   64
#define H_   128
#define W_   2048
#define HW_  (H_ * W_)      // 262144 = 2^18
#define NPIX (B_ * HW_)     // 16,777,216

#define PI_F 3.14159265358979f

// ---------------------------------------------------------------------------
// Per-batch augmentation parameters (one 128B struct per batch image).
// Lives at the start of d_ws; the 67MB intermediate image follows at +8KB.
// ---------------------------------------------------------------------------
struct Params {
    float a11, a12, a21, a22, txt, tyt, geo;   // inverse affine + geometric mask
    float wamp, wfreq, warp;                   // page-warp amplitude/freq + mask
    float eros, ero;                           // erosion strength + mask
    float thr, binz;                           // binarize threshold + mask
    float nscale, jpeg;                        // jpeg-noise scale + mask
    float br, co, cang, sang, light;           // lighting params + mask
    float pad[11];                             // pad to 32 floats (128 B)
};

// ---------------------------------------------------------------------------
// Counter-based RNG (PCG-style hash): deterministic, stateless.
// ---------------------------------------------------------------------------
__device__ __forceinline__ uint32_t pcg_hash(uint32_t v) {
    v = v * 747796405u + 2891336453u;
    uint32_t w = ((v >> ((v >> 28u) + 4u)) ^ v) * 277803737u;
    return (w >> 22u) ^ w;
}
__device__ __forceinline__ float u01(uint32_t a, uint32_t b) {
    uint32_t h = pcg_hash(a * 0x9E3779B9u ^ (b + 0x7F4A7C15u));
    return (float)(h >> 8) * (1.0f / 16777216.0f);   // [0,1)
}

// ---------------------------------------------------------------------------
// Kernel 0: generate per-batch random parameters (64 threads total).
// ---------------------------------------------------------------------------
__global__ void aug_params_kernel(Params* __restrict__ P) {
    int b = threadIdx.x;
    if (b >= B_) return;
    uint32_t s = (uint32_t)b;

    float zoom = __expf(u01(s, 0) * 0.2f - 0.1f);          // exp(U[-.1,.1])
    float ty   = (u01(s, 1) * 2.f - 1.f) * (0.05f * H_);   // U[-.05H, .05H]
    float tx   = -u01(s, 2) * (0.05f * W_);                // U[-.05W, 0]
    float shx  = (u01(s, 3) * 2.f - 1.f) * 0.1f;
    float shy  = (u01(s, 4) * 2.f - 1.f) * 0.1f;
    float cy   = 0.5f * (float)H_;

    Params p;
    p.a11 = zoom + shx;  p.a12 = shx;
    p.a21 = shy;         p.a22 = zoom + shy;
    p.txt = cy * (1.f - p.a22) + tx;
    p.tyt = cy * (1.f - p.a22) + ty;
    p.geo = (u01(s, 5) < 0.80f) ? 1.f : 0.f;

    p.wamp  = u01(s, 6) * 0.02f;
    p.wfreq = 0.5f + u01(s, 7) * 1.5f;
    p.warp  = (u01(s, 8) < 0.25f) ? 1.f : 0.f;

    p.eros = u01(s, 9) * 0.5f;
    p.ero  = (u01(s, 10) < 0.25f) ? 1.f : 0.f;

    p.thr  = 0.3f + u01(s, 11) * 0.4f;
    p.binz = (u01(s, 12) < 0.25f) ? 1.f : 0.f;

    float q  = 30.f + u01(s, 13) * 60.f;
    p.nscale = (100.f - q) * (0.1f / 100.f);
    p.jpeg   = (u01(s, 15) < 0.125f) ? 1.f : 0.f;

    p.br = (u01(s, 16) * 2.f - 1.f) * 0.2f;
    p.co = 0.8f + u01(s, 17) * 0.4f;
    float ang = u01(s, 18) * 6.28318530718f;
    p.cang  = __cosf(ang);
    p.sang  = __sinf(ang);
    p.light = (u01(s, 19) < 0.25f) ? 1.f : 0.f;

    for (int i = 0; i < 11; ++i) p.pad[i] = 0.f;
    P[b] = p;
}

// ---------------------------------------------------------------------------
// Kernel 1: inverse-affine bilinear resample (constant fill), blended with the
// original by the per-batch geometric mask. Streams into workspace X.
// 1 pixel / thread; a block covers 256 contiguous pixels of one row, so all
// Params loads are wave-uniform (scalarizable).
// ---------------------------------------------------------------------------
__global__ __launch_bounds__(256)
void aug_geo_kernel(const float* __restrict__ in, float* __restrict__ X,
                    const Params* __restrict__ P) {
    int idx = blockIdx.x * blockDim.x + threadIdx.x;
    int b   = idx >> 18;          // / HW_
    int rem = idx & (HW_ - 1);
    int y   = rem >> 11;          // / W_
    int x   = rem & (W_ - 1);

    const Params& p = P[b];
    float orig = in[idx];
    float out  = orig;

    if (p.geo != 0.f) {
        float xf = (float)x, yf = (float)y;
        float xin = fmaf(p.a11, xf, fmaf(p.a12, yf, p.txt));
        float yin = fmaf(p.a21, xf, fmaf(p.a22, yf, p.tyt));
        float x0f = floorf(xin), y0f = floorf(yin);
        int   x0  = (int)x0f,    y0  = (int)y0f;
        float fx  = xin - x0f,   fy  = yin - y0f;
        const float* base = in + (size_t)b * HW_;

        auto tap = [&](int yy, int xx) -> float {
            bool valid = (yy >= 0) & (yy < H_) & (xx >= 0) & (xx < W_);
            int yc = min(max(yy, 0), H_ - 1);
            int xc = min(max(xx, 0), W_ - 1);
            float v = base[yc * W_ + xc];
            return valid ? v : 0.f;                 // CONSTANT fill (0)
        };
        float I00 = tap(y0,     x0), I01 = tap(y0,     x0 + 1);
        float I10 = tap(y0 + 1, x0), I11 = tap(y0 + 1, x0 + 1);
        float I0  = fmaf(I01 - I00, fx, I00);
        float I1  = fmaf(I11 - I10, fx, I10);
        out       = fmaf(I1 - I0, fy, I0);
    }
    X[idx] = out;
}

// ---------------------------------------------------------------------------
// Kernel 2: page-warp (bilinear, edge-clamped) + erosion + binarize + jpeg
// noise + lighting + clip, fused. One block per (batch, row).
//
// |disp_y| <= 0.02*H*0.3 = 0.77 px -> warp samples for row y touch only rows
// y-1..y+1 of the intermediate. Those 3 rows (24 KB) are staged into LDS with
// CDNA5 async global->LDS b128 loads (ASYNCcnt path), then the bilinear warp
// becomes an LDS gather. disp_x is row-uniform and hoisted out of the loop.
// ---------------------------------------------------------------------------
__global__ __launch_bounds__(256)
void aug_warp_post_kernel(const float* __restrict__ X, float* __restrict__ out,
                          const Params* __restrict__ P) {
    __shared__ float rows[3 * W_];          // slot0=y-1, slot1=y, slot2=y+1 (clamped)

    int bid = blockIdx.x;
    int b   = bid >> 7;                     // / H_
    int y   = bid & (H_ - 1);
    int t   = threadIdx.x;
    const Params& p = P[b];

    int yA = max(y - 1, 0);
    int yC = min(y + 1, H_ - 1);
    const float* src0 = X + (size_t)(b * H_ + yA) * W_;
    const float* src1 = X + (size_t)(b * H_ + y ) * W_;
    const float* src2 = X + (size_t)(b * H_ + yC) * W_;

    // ---- async stage: 512 float4 chunks per row, 256 threads -> 2 chunks each.
    uint32_t lds_base = (uint32_t)(uintptr_t)(void*)&rows[0];   // low 32b = LDS offset
#pragma unroll
    for (int i = 0; i < 2; ++i) {
        int c = t + 256 * i;                       // float4 chunk within the row
        uint32_t loff = lds_base + (uint32_t)(c * 16);
        const float* g0 = src0 + c * 4;
        const float* g1 = src1 + c * 4;
        const float* g2 = src2 + c * 4;
        asm volatile("global_load_async_to_lds_b128 %0, %1, off"
                     :: "v"(loff),                              "v"(g0) : "memory");
        asm volatile("global_load_async_to_lds_b128 %0, %1, off"
                     :: "v"(loff + (uint32_t)(W_ * 4)),         "v"(g1) : "memory");
        asm volatile("global_load_async_to_lds_b128 %0, %1, off"
                     :: "v"(loff + (uint32_t)(2 * W_ * 4)),     "v"(g2) : "memory");
    }
    asm volatile("s_wait_asynccnt 0" ::: "memory");
    __syncthreads();

    // Row-uniform warp terms.
    bool  dowarp = (p.warp != 0.f);
    float dx = p.wamp * (float)W_ * __sinf(p.wfreq * PI_F * (float)y * (1.0f / H_));
    float yl = (float)y * (2.0f / (H_ - 1)) - 1.0f;
    float co = p.co, br = p.br;

    float* orow = out + (size_t)(b * H_ + y) * W_;

#pragma unroll
    for (int j = 0; j < 8; ++j) {
        int x = t + 256 * j;                // lane-consecutive -> stride-1 LDS banks
        float v;

        if (dowarp) {
            float xf = (float)x;
            float dy = p.wamp * (float)H_ * 0.3f *
                       __sinf(p.wfreq * PI_F * xf * (1.0f / W_));
            float nx = fminf(fmaxf(xf + dx, 0.f), (float)W_ - 1.001f);
            float ny = fminf(fmaxf((float)y + dy, 0.f), (float)H_ - 1.001f);
            float x0f = floorf(nx), y0f = floorf(ny);
            int   x0  = (int)x0f,   y0  = (int)y0f;
            float fx  = nx - x0f,   fy  = ny - y0f;
            int y1 = min(y0 + 1, H_ - 1);
            int s0 = min(max(y0 - (y - 1), 0), 2);   // LDS slot of row y0
            int s1 = min(max(y1 - (y - 1), 0), 2);   // LDS slot of row y1
            int x1 = min(x0 + 1, W_ - 1);
            float I00 = rows[s0 * W_ + x0], I01 = rows[s0 * W_ + x1];
            float I10 = rows[s1 * W_ + x0], I11 = rows[s1 * W_ + x1];
            float I0  = fmaf(I01 - I00, fx, I00);
            float I1  = fmaf(I11 - I10, fx, I10);
            v = fmaf(I1 - I0, fy, I0);
        } else {
            v = rows[W_ + x];               // slot1 = row y of the intermediate
        }

        // erosion (1x1 erosion == x-1): x*(1-s)+(x-1)*s == x - s
        v = (p.ero != 0.f) ? (v - p.eros) : v;

        // binarize
        v = (p.binz != 0.f) ? ((v > p.thr) ? 1.f : 0.f) : v;

        // jpeg-artifact noise: per-pixel Gaussian via Box-Muller
        if (p.jpeg != 0.f) {
            uint32_t pix = (uint32_t)((b << 18) | (y << 11) | x);
            uint32_t h1  = pcg_hash(pix ^ 0xB5297A4Du);
            uint32_t h2  = pcg_hash(h1  ^ 0x68E31DA4u);
            float u1 = (float)((h1 >> 8) + 1u) * (1.0f / 16777216.0f);
            float u2 = (float)(h2 >> 8)        * (1.0f / 16777216.0f);
            float g  = __fsqrt_rn(-2.f * __logf(u1)) * __cosf(6.28318530718f * u2);
            v = fmaf(g, p.nscale, v);
        }

        // spatially-varying lighting
        if (p.light != 0.f) {
            float xl   = (float)x * (2.0f / (W_ - 1)) - 1.0f;
            float grad = fmaf(p.cang, xl, p.sang * yl);
            v = fmaf(v - 0.5f, co, fmaf(br, grad, 0.5f));
        }

        orow[x] = fminf(fmaxf(v, 0.f), 1.f);
    }
}

// ---------------------------------------------------------------------------
// Host-side launcher. d_ws layout: [0, 8KB) Params[64]; [8KB, 8KB+67MB) X.
// ---------------------------------------------------------------------------
extern "C" void kernel_launch(void* const* d_in, const int* in_sizes, int n_in,
                              void* d_out, int out_size, void* d_ws, size_t ws_size,
                              hipStream_t stream) {
    (void)in_sizes; (void)n_in; (void)out_size; (void)ws_size;

    const float* images = (const float*)d_in[0];
    float*       outp   = (float*)d_out;
    Params*      P      = (Params*)d_ws;
    float*       X      = (float*)((char*)d_ws + 8192);

    aug_params_kernel<<<1, 64, 0, stream>>>(P);
    aug_geo_kernel<<<NPIX / 256, 256, 0, stream>>>(images, X, P);
    aug_warp_post_kernel<<<B_ * H_, 256, 0, stream>>>(X, outp, P);
}